// GAT_29222957482263
// MI455X (gfx1250) — compile-verified
//
#include <hip/hip_runtime.h>
#include <hip/hip_bf16.h>
#include <math.h>

#define IN_CH 128
#define HEADS 4
#define HID   32
#define NCLS  10
#define NEG_SLOPE 0.2f

typedef float v2f __attribute__((ext_vector_type(2)));
typedef float v8f __attribute__((ext_vector_type(8)));

__device__ __forceinline__ float leaky(float v) { return v > 0.f ? v : NEG_SLOPE * v; }

// float atomic max via int/uint atomics (valid for mixed signs, init = -3e38)
__device__ __forceinline__ void atomicMaxF(float* addr, float val) {
    if (val >= 0.f) atomicMax((int*)addr, __float_as_int(val));
    else            atomicMin((unsigned int*)addr, __float_as_uint(val));
}

// ---------------- init ----------------
__global__ void gat_init(float* m1, float* den1, float* acc1,
                         float* m2, float* den2, float* acc2, int n) {
    long long i = blockIdx.x * (long long)blockDim.x + threadIdx.x;
    if (i < (long long)n * IN_CH) acc1[i] = 0.f;
    if (i < (long long)n * HEADS) { m1[i] = -3.0e38f; den1[i] = 0.f; }
    if (i < n)                    { m2[i] = -3.0e38f; den2[i] = 0.f; }
    if (i < (long long)n * NCLS)  acc2[i] = 0.f;
}

// ---------------- GEMM1: H[n,128] = X[n,128] @ W[128,128], f32 WMMA ----------------
// W staged once per block into LDS, k-pairs packed adjacent so each B fragment
// is a single ds_load_b64. 8 waves/block, each wave owns a 16-row x 128-col strip.
__global__ void __launch_bounds__(256) gemm1_wmma(const float* __restrict__ X,
                                                  const float* __restrict__ W,
                                                  float* __restrict__ H, int n) {
    __shared__ float2 wl[64 * 128];   // 64 KB: [g][col], g encodes (k0/4, khalf)
    int tid = threadIdx.x;
    for (int i = tid; i < 64 * 128; i += 256) {
        int g = i >> 7, col = i & 127;
        int k = (g >> 1) * 4 + (g & 1) * 2;   // g = (k0/4)*2 + khalf -> k = k0 + 2*khalf
        wl[i] = make_float2(W[(size_t)k * 128 + col], W[(size_t)(k + 1) * 128 + col]);
    }
    __syncthreads();

    int wave = blockIdx.x * 8 + (tid >> 5);
    int lane = tid & 31;
    int ntiles = (n + 15) >> 4;
    if (wave >= ntiles) return;
    int rloc  = lane & 15;     // row within tile (A), col within tile (B/C)
    int khalf = lane >> 4;     // 0 -> k pair {0,1}, 1 -> k pair {2,3}
    int row = wave * 16 + rloc;
    if (row >= n) row = n - 1;
    v8f acc[8] = {};
    const float* xrow = X + (size_t)row * IN_CH;
    for (int k0 = 0; k0 < IN_CH; k0 += 4) {
        int k = k0 + 2 * khalf;
        v2f a; a.x = xrow[k]; a.y = xrow[k + 1];
        const float2* bl = &wl[((k0 >> 2) * 2 + khalf) * 128];
        #pragma unroll
        for (int t = 0; t < 8; ++t) {
            float2 bp = bl[t * 16 + rloc];
            v2f b; b.x = bp.x; b.y = bp.y;
            acc[t] = __builtin_amdgcn_wmma_f32_16x16x4_f32(
                false, a, false, b, (short)0, acc[t], false, false);
        }
    }
    #pragma unroll
    for (int t = 0; t < 8; ++t) {
        #pragma unroll
        for (int r = 0; r < 8; ++r) {
            int m = wave * 16 + khalf * 8 + r;   // C layout: vgpr r = row r (lanes 0-15) / r+8 (16-31)
            if (m < n) H[(size_t)m * 128 + t * 16 + rloc] = acc[t][r];
        }
    }
}

// ---------------- GEMM2: H2[n,16] = X[n,128] @ pad16(W2[128,10]) ----------------
__global__ void __launch_bounds__(256) gemm2_wmma(const float* __restrict__ X,
                                                  const float* __restrict__ W2,
                                                  float* __restrict__ H2, int n) {
    __shared__ float2 wl[64 * 16];    // 8 KB, zero-padded to 16 cols
    int tid = threadIdx.x;
    for (int i = tid; i < 64 * 16; i += 256) {
        int g = i >> 4, col = i & 15;
        int k = (g >> 1) * 4 + (g & 1) * 2;
        float b0 = (col < NCLS) ? W2[(size_t)k * NCLS + col]       : 0.f;
        float b1 = (col < NCLS) ? W2[(size_t)(k + 1) * NCLS + col] : 0.f;
        wl[i] = make_float2(b0, b1);
    }
    __syncthreads();

    int wave = blockIdx.x * 8 + (tid >> 5);
    int lane = tid & 31;
    int ntiles = (n + 15) >> 4;
    if (wave >= ntiles) return;
    int rloc  = lane & 15;
    int khalf = lane >> 4;
    int row = wave * 16 + rloc;
    if (row >= n) row = n - 1;
    v8f acc = {};
    const float* xrow = X + (size_t)row * IN_CH;
    for (int k0 = 0; k0 < IN_CH; k0 += 4) {
        int k = k0 + 2 * khalf;
        v2f a; a.x = xrow[k]; a.y = xrow[k + 1];
        float2 bp = wl[((k0 >> 2) * 2 + khalf) * 16 + rloc];
        v2f b; b.x = bp.x; b.y = bp.y;
        acc = __builtin_amdgcn_wmma_f32_16x16x4_f32(
            false, a, false, b, (short)0, acc, false, false);
    }
    #pragma unroll
    for (int r = 0; r < 8; ++r) {
        int m = wave * 16 + khalf * 8 + r;
        if (m < n) H2[(size_t)m * 16 + rloc] = acc[r];
    }
}

// ---------------- attention dot products ----------------
__global__ void att_dots1(const float* __restrict__ H, const float* __restrict__ asw,
                          const float* __restrict__ adw, float* a_s, float* a_d, int n) {
    long long i = blockIdx.x * (long long)blockDim.x + threadIdx.x;
    if (i >= (long long)n * HEADS) return;
    int node = (int)(i >> 2), h = (int)(i & 3);
    const float* hp = H + (size_t)node * IN_CH + h * HID;
    float s = 0.f, d = 0.f;
    #pragma unroll
    for (int c = 0; c < HID; ++c) { float v = hp[c]; s += v * asw[h * HID + c]; d += v * adw[h * HID + c]; }
    a_s[i] = s; a_d[i] = d;
}

__global__ void att_dots2(const float* __restrict__ H2, const float* __restrict__ asw,
                          const float* __restrict__ adw, float* a_s, float* a_d, int n) {
    int i = blockIdx.x * blockDim.x + threadIdx.x;
    if (i >= n) return;
    const float* hp = H2 + (size_t)i * 16;
    float s = 0.f, d = 0.f;
    #pragma unroll
    for (int c = 0; c < NCLS; ++c) { float v = hp[c]; s += v * asw[c]; d += v * adw[c]; }
    a_s[i] = s; a_d[i] = d;
}

// ---------------- layer 1 edge passes ----------------
__global__ void edge_max1(const long long* __restrict__ ei, long long E, int n,
                          const float* __restrict__ a_s, const float* __restrict__ a_d,
                          float* m1) {
    long long i = blockIdx.x * (long long)blockDim.x + threadIdx.x;
    if (i >= (E + n) * HEADS) return;
    long long e = i >> 2; int h = (int)(i & 3);
    int s, d;
    if (e < E) { s = (int)ei[e]; d = (int)ei[E + e]; } else { s = d = (int)(e - E); }
    float v = leaky(a_s[(size_t)s * 4 + h] + a_d[(size_t)d * 4 + h]);
    atomicMaxF(&m1[(size_t)d * 4 + h], v);
}

// one wave per edge; lane = channel; accumulate denom and un-normalized numerator
__global__ void edge_agg1(const long long* __restrict__ ei, long long E, int n,
                          const float* __restrict__ a_s, const float* __restrict__ a_d,
                          const float* __restrict__ m1, const float* __restrict__ H,
                          float* __restrict__ den1, float* __restrict__ acc1) {
    long long tid = blockIdx.x * (long long)blockDim.x + threadIdx.x;
    int lane = threadIdx.x & 31;
    long long e = tid >> 5;
    if (e >= E + n) return;
    int s, d;
    if (e < E) { s = (int)ei[e]; d = (int)ei[E + e]; } else { s = d = (int)(e - E); }
    float4 as4 = *reinterpret_cast<const float4*>(a_s + 4 * (size_t)s);
    float4 ad4 = *reinterpret_cast<const float4*>(a_d + 4 * (size_t)d);
    float4 mm4 = *reinterpret_cast<const float4*>(m1  + 4 * (size_t)d);
    float ex[4];
    ex[0] = __expf(leaky(as4.x + ad4.x) - mm4.x);
    ex[1] = __expf(leaky(as4.y + ad4.y) - mm4.y);
    ex[2] = __expf(leaky(as4.z + ad4.z) - mm4.z);
    ex[3] = __expf(leaky(as4.w + ad4.w) - mm4.w);
    if (lane < 4) atomicAdd(&den1[4 * (size_t)d + lane], ex[lane]);
    const float* hs = H + (size_t)s * IN_CH;
    float* ac = acc1 + (size_t)d * IN_CH;
    #pragma unroll
    for (int h = 0; h < HEADS; ++h)
        atomicAdd(&ac[h * HID + lane], ex[h] * hs[h * HID + lane]);
}

// normalize + bias + ELU, in place (output feeds GEMM2)
__global__ void norm_elu1(float* acc1, const float* __restrict__ den1,
                          const float* __restrict__ b1, int n) {
    long long i = blockIdx.x * (long long)blockDim.x + threadIdx.x;
    if (i >= (long long)n * IN_CH) return;
    int node = (int)(i >> 7);
    int f = (int)(i & 127);
    float v = acc1[i] / den1[(size_t)node * 4 + (f >> 5)] + b1[f];
    acc1[i] = v > 0.f ? v : (__expf(v) - 1.f);
}

// ---------------- layer 2 edge passes ----------------
__global__ void edge_max2(const long long* __restrict__ ei, long long E, int n,
                          const float* __restrict__ a_s, const float* __restrict__ a_d,
                          float* m2) {
    long long e = blockIdx.x * (long long)blockDim.x + threadIdx.x;
    if (e >= E + n) return;
    int s, d;
    if (e < E) { s = (int)ei[e]; d = (int)ei[E + e]; } else { s = d = (int)(e - E); }
    atomicMaxF(&m2[d], leaky(a_s[s] + a_d[d]));
}

__global__ void edge_agg2(const long long* __restrict__ ei, long long E, int n,
                          const float* __restrict__ a_s, const float* __restrict__ a_d,
                          const float* __restrict__ m2, const float* __restrict__ H2,
                          float* __restrict__ den2, float* __restrict__ acc2) {
    long long e = blockIdx.x * (long long)blockDim.x + threadIdx.x;
    if (e >= E + n) return;
    int s, d;
    if (e < E) { s = (int)ei[e]; d = (int)ei[E + e]; } else { s = d = (int)(e - E); }
    float ex = __expf(leaky(a_s[s] + a_d[d]) - m2[d]);
    atomicAdd(&den2[d], ex);
    const float* hs = H2 + (size_t)s * 16;
    float* ac = acc2 + (size_t)d * NCLS;
    #pragma unroll
    for (int c = 0; c < NCLS; ++c) atomicAdd(&ac[c], ex * hs[c]);
}

__global__ void final_out(const float* __restrict__ acc2, const float* __restrict__ den2,
                          const float* __restrict__ b2, float* __restrict__ out, int n) {
    long long i = blockIdx.x * (long long)blockDim.x + threadIdx.x;
    if (i >= (long long)n * NCLS) return;
    int node = (int)(i / NCLS), c = (int)(i % NCLS);
    out[i] = acc2[i] / den2[node] + b2[c];
}

extern "C" void kernel_launch(void* const* d_in, const int* in_sizes, int n_in,
                              void* d_out, int out_size, void* d_ws, size_t ws_size,
                              hipStream_t stream) {
    const float* x         = (const float*)d_in[0];
    const long long* ei    = (const long long*)d_in[1];
    const float* W1        = (const float*)d_in[2];
    const float* att_src1  = (const float*)d_in[3];
    const float* att_dst1  = (const float*)d_in[4];
    const float* b1        = (const float*)d_in[5];
    const float* W2        = (const float*)d_in[6];
    const float* att_src2  = (const float*)d_in[7];
    const float* att_dst2  = (const float*)d_in[8];
    const float* b2        = (const float*)d_in[9];
    int n = in_sizes[0] / IN_CH;
    long long E = in_sizes[1] / 2;
    float* out = (float*)d_out;

    float* ws = (float*)d_ws;
    size_t off = 0;
    float* h1   = ws + off; off += (size_t)n * IN_CH;
    float* acc1 = ws + off; off += (size_t)n * IN_CH;
    float* a_s1 = ws + off; off += (size_t)n * HEADS;
    float* a_d1 = ws + off; off += (size_t)n * HEADS;
    float* m1   = ws + off; off += (size_t)n * HEADS;
    float* den1 = ws + off; off += (size_t)n * HEADS;
    float* h2   = ws + off; off += (size_t)n * 16;
    float* a_s2 = ws + off; off += (size_t)n;
    float* a_d2 = ws + off; off += (size_t)n;
    float* m2   = ws + off; off += (size_t)n;
    float* den2 = ws + off; off += (size_t)n;
    float* acc2 = ws + off; off += (size_t)n * NCLS;

    long long Etot = E + n;
    const int B = 256;
    auto nb = [](long long t, int b) { return (unsigned)((t + b - 1) / b); };
    int tiles = (n + 15) / 16;
    unsigned gemm_blocks = (unsigned)((tiles + 7) / 8);   // 8 waves (16-row tiles) per block

    gat_init  <<<nb((long long)n * IN_CH, B), B, 0, stream>>>(m1, den1, acc1, m2, den2, acc2, n);
    gemm1_wmma<<<gemm_blocks, B, 0, stream>>>(x, W1, h1, n);
    att_dots1 <<<nb((long long)n * HEADS, B), B, 0, stream>>>(h1, att_src1, att_dst1, a_s1, a_d1, n);
    edge_max1 <<<nb(Etot * HEADS, B), B, 0, stream>>>(ei, E, n, a_s1, a_d1, m1);
    edge_agg1 <<<nb(Etot * 32, B), B, 0, stream>>>(ei, E, n, a_s1, a_d1, m1, h1, den1, acc1);
    norm_elu1 <<<nb((long long)n * IN_CH, B), B, 0, stream>>>(acc1, den1, b1, n);
    gemm2_wmma<<<gemm_blocks, B, 0, stream>>>(acc1, W2, h2, n);
    att_dots2 <<<nb(n, B), B, 0, stream>>>(h2, att_src2, att_dst2, a_s2, a_d2, n);
    edge_max2 <<<nb(Etot, B), B, 0, stream>>>(ei, E, n, a_s2, a_d2, m2);
    edge_agg2 <<<nb(Etot, B), B, 0, stream>>>(ei, E, n, a_s2, a_d2, m2, h2, den2, acc2);
    final_out <<<nb((long long)n * NCLS, B), B, 0, stream>>>(acc2, den2, b2, out, n);
}